// MeshGNN_16003048145307
// MI455X (gfx1250) — compile-verified
//
#include <hip/hip_runtime.h>
#include <hip/hip_bf16.h>
#include <math.h>

#define N_NODES 100000
#define HID 64

typedef __attribute__((ext_vector_type(2))) float v2f;
typedef __attribute__((ext_vector_type(8))) float v8f;

// ---------------- utility: zero fill (grid-stride) ----------------
__global__ void fill_zero_kernel(float* __restrict__ p, int n) {
    int i = blockIdx.x * blockDim.x + threadIdx.x;
    int stride = gridDim.x * blockDim.x;
    for (; i < n; i += stride) p[i] = 0.0f;
}

// ---------------- degree count via atomics ----------------
__global__ void degree_kernel(const int* __restrict__ col, int E,
                              float* __restrict__ deg) {
    int e = blockIdx.x * blockDim.x + threadIdx.x;
    if (e < E) atomicAdd(&deg[col[e]], 1.0f);
}

// deg_inv_sqrt = rsqrt(deg + 1)   (the +1 is the self-loop; always > 0)
__global__ void rsqrt_kernel(float* __restrict__ deg, int n) {
    int i = blockIdx.x * blockDim.x + threadIdx.x;
    if (i < n) deg[i] = rsqrtf(deg[i] + 1.0f);
}

// ---------------- WMMA fp32 GEMM: out[M,64] = A[M,K] @ W[K,64] ----------------
// One wave computes a 16(M) x 64(N) tile: 4 accumulators of v8f,
// V_WMMA_F32_16X16X4_F32 stepping K by 4.
__global__ __launch_bounds__(256) void gemm_n64_kernel(
    const float* __restrict__ A, const float* __restrict__ W,
    float* __restrict__ out, int M, int K) {
    const int wave = threadIdx.x >> 5;
    const int lane = threadIdx.x & 31;
    const int m0 = (blockIdx.x * 8 + wave) * 16;
    if (m0 >= M) return;                 // whole-wave exit: EXEC all-1s for WMMA

    const int lrow  = lane & 15;         // A: row M, B: col N, C/D: col N
    const int khalf = (lane >> 4) * 2;   // K sub-pair per half-wave

    v8f acc0 = {}, acc1 = {}, acc2 = {}, acc3 = {};

    for (int k = 0; k < K; k += 4) {
        // A fragment: 16x4 fp32. lane holds A[m0+lrow][k+khalf .. +1]
        const float* ap = A + (size_t)(m0 + lrow) * K + (k + khalf);
        v2f a; a.x = ap[0]; a.y = ap[1];

        // B fragments: 4x16 fp32 per n-tile. lane holds W[k+khalf..+1][n]
        const float* wp = W + (size_t)(k + khalf) * HID + lrow;
        v2f b0, b1, b2, b3;
        b0.x = wp[0];   b0.y = wp[HID];
        b1.x = wp[16];  b1.y = wp[16 + HID];
        b2.x = wp[32];  b2.y = wp[32 + HID];
        b3.x = wp[48];  b3.y = wp[48 + HID];

        acc0 = __builtin_amdgcn_wmma_f32_16x16x4_f32(false, a, false, b0,
                                                     (short)0, acc0, false, false);
        acc1 = __builtin_amdgcn_wmma_f32_16x16x4_f32(false, a, false, b1,
                                                     (short)0, acc1, false, false);
        acc2 = __builtin_amdgcn_wmma_f32_16x16x4_f32(false, a, false, b2,
                                                     (short)0, acc2, false, false);
        acc3 = __builtin_amdgcn_wmma_f32_16x16x4_f32(false, a, false, b3,
                                                     (short)0, acc3, false, false);
    }

    // C/D layout: VGPR v -> row m0 + v + 8*(lane>>4), col = lrow (+ ntile*16)
    float* op = out + (size_t)(m0 + (lane >> 4) * 8) * HID + lrow;
    #pragma unroll
    for (int v = 0; v < 8; ++v) {
        op[(size_t)v * HID + 0]  = acc0[v];
        op[(size_t)v * HID + 16] = acc1[v];
        op[(size_t)v * HID + 32] = acc2[v];
        op[(size_t)v * HID + 48] = acc3[v];
    }
}

// ---------------- edge scatter-add: 64 lanes per edge (coalesced) ----------------
// out[col[e], f] += xw[row[e], f] * dis[row[e]] * dis[col[e]]
__global__ __launch_bounds__(256) void scatter_kernel(
    const float* __restrict__ xw, const int* __restrict__ row,
    const int* __restrict__ col, const float* __restrict__ dis,
    float* __restrict__ out, int E) {
    int gid = blockIdx.x * 256 + threadIdx.x;
    int e = gid >> 6;
    int f = gid & 63;
    if (e >= E) return;
    int r = row[e];
    int c = col[e];
    float norm = dis[r] * dis[c];
    atomicAdd(&out[(size_t)c * HID + f], xw[(size_t)r * HID + f] * norm);
}

// ---------------- epilogue: self-loop + bias + ReLU (in place) ----------------
// agg[n,f] = relu(agg[n,f] + xw[n,f]*dis[n]^2 + b[f])
__global__ __launch_bounds__(256) void epilogue_kernel(
    float* __restrict__ agg, const float* __restrict__ xw,
    const float* __restrict__ dis, const float* __restrict__ b, int n64) {
    int i = blockIdx.x * 256 + threadIdx.x;
    if (i >= n64) return;
    int n = i >> 6;
    int f = i & 63;
    float d = dis[n];
    float v = agg[i] + xw[i] * d * d + b[f];
    agg[i] = v > 0.0f ? v : 0.0f;
}

// ---------------- final: sigmoid(h @ Wf + bf), one wave per node ----------------
__global__ __launch_bounds__(256) void final_kernel(
    const float* __restrict__ h, const float* __restrict__ Wf,
    const float* __restrict__ bf, float* __restrict__ out, int N) {
    int wave = (blockIdx.x * 256 + threadIdx.x) >> 5;
    int lane = threadIdx.x & 31;
    if (wave >= N) return;
    const float* hp = h + (size_t)wave * HID;
    float s = hp[lane] * Wf[lane] + hp[lane + 32] * Wf[lane + 32];
    #pragma unroll
    for (int off = 16; off > 0; off >>= 1) s += __shfl_down(s, off, 32);
    if (lane == 0) {
        float z = s + bf[0];
        out[wave] = 1.0f / (1.0f + expf(-z));
    }
}

extern "C" void kernel_launch(void* const* d_in, const int* in_sizes, int n_in,
                              void* d_out, int out_size, void* d_ws, size_t ws_size,
                              hipStream_t stream) {
    const float* x  = (const float*)d_in[0];   // [N,16]
    const int*   ei = (const int*)d_in[1];     // [2,E]
    const float* W1 = (const float*)d_in[2];   // [16,64]
    const float* b1 = (const float*)d_in[3];   // [64]
    const float* W2 = (const float*)d_in[4];   // [64,64]
    const float* b2 = (const float*)d_in[5];   // [64]
    const float* Wf = (const float*)d_in[6];   // [64,1]
    const float* bf = (const float*)d_in[7];   // [1]
    float* out = (float*)d_out;

    const int N = N_NODES;
    const int E = in_sizes[1] / 2;
    const int* row = ei;       // edge_index[0] : sources
    const int* col = ei + E;   // edge_index[1] : targets

    // workspace layout
    float* dis  = (float*)d_ws;                        // [N] deg_inv_sqrt
    float* bufA = dis + 100096;                        // [N,64] pre-aggregation feats
    float* bufB = bufA + (size_t)N * HID;              // [N,64] aggregation target

    const int n64 = N * HID;

    // --- degree / normalization ---
    fill_zero_kernel<<<1024, 256, 0, stream>>>(dis, N);
    degree_kernel<<<(E + 255) / 256, 256, 0, stream>>>(col, E, dis);
    rsqrt_kernel<<<(N + 255) / 256, 256, 0, stream>>>(dis, N);

    const int gemm_blocks = (N + 127) / 128;     // 8 waves/block, 16 rows/wave
    const int sc_blocks   = (E * 64 + 255) / 256;
    const int ep_blocks   = (n64 + 255) / 256;

    // --- layer 1 ---
    gemm_n64_kernel<<<gemm_blocks, 256, 0, stream>>>(x, W1, bufA, N, 16);
    fill_zero_kernel<<<4096, 256, 0, stream>>>(bufB, n64);
    scatter_kernel<<<sc_blocks, 256, 0, stream>>>(bufA, row, col, dis, bufB, E);
    epilogue_kernel<<<ep_blocks, 256, 0, stream>>>(bufB, bufA, dis, b1, n64);

    // --- layer 2 (bufB holds h1; consumed by GEMM, then reused as agg target) ---
    gemm_n64_kernel<<<gemm_blocks, 256, 0, stream>>>(bufB, W2, bufA, N, 64);
    fill_zero_kernel<<<4096, 256, 0, stream>>>(bufB, n64);
    scatter_kernel<<<sc_blocks, 256, 0, stream>>>(bufA, row, col, dis, bufB, E);
    epilogue_kernel<<<ep_blocks, 256, 0, stream>>>(bufB, bufA, dis, b2, n64);

    // --- readout ---
    final_kernel<<<(N * 32 + 255) / 256, 256, 0, stream>>>(bufB, Wf, bf, out, N);
}